// GNNML3_Structural_74577812128608
// MI455X (gfx1250) — compile-verified
//
#include <hip/hip_runtime.h>
#include <hip/hip_bf16.h>
#include <cmath>

#define N_GRAPHS 2000
#define NEURON   48
#define CLOSE_DIM 8
#define HID      16
#define BLK      384   // 12 waves; 384 = 8 nodes x 48 cols
#define TILE     64

typedef __attribute__((ext_vector_type(2))) float v2f;
typedef __attribute__((ext_vector_type(8))) float v8f;

// ---------------------------------------------------------------------------
// Kernel 1: segment boundaries via binary search (batch is sorted ascending)
// start[g] = first index i with batch[i] >= g ; start[G] = N
// ---------------------------------------------------------------------------
__global__ void seg_bounds_kernel(const int* __restrict__ batch, int n_nodes,
                                  int* __restrict__ start) {
  int g = blockIdx.x * blockDim.x + threadIdx.x;
  if (g > N_GRAPHS) return;
  if (g == N_GRAPHS) { start[g] = n_nodes; return; }
  int lo = 0, hi = n_nodes;
  while (lo < hi) {
    int mid = (lo + hi) >> 1;
    if (batch[mid] < g) lo = mid + 1; else hi = mid;
  }
  start[g] = lo;
}

// ---------------------------------------------------------------------------
// Kernel 2: per-graph attention pooling, one block per graph.
// pooled[g][c] = (n/den) * max_i exp(s_i - smax) * x[i][c]
// ---------------------------------------------------------------------------
__global__ __launch_bounds__(BLK)
void attn_pool_kernel(const float* __restrict__ x,
                      const float* __restrict__ close,
                      const int*   __restrict__ start,
                      const float* __restrict__ Wc,
                      const float* __restrict__ bc,
                      float* __restrict__ pooled) {
  __shared__ float red[512];
  __shared__ float etile[TILE];

  const int tid = threadIdx.x;
  const int g   = blockIdx.x;
  const int i0  = start[g];
  const int i1  = start[g + 1];
  const int n   = i1 - i0;

  float wc[CLOSE_DIM];
#pragma unroll
  for (int k = 0; k < CLOSE_DIM; ++k) wc[k] = Wc[k];
  const float bias = bc[0];

  if (n <= 0) {  // never happens statistically; keep output defined
    if (tid < NEURON) pooled[g * NEURON + tid] = 0.0f;
    return;
  }

  // ---- pass 1: per-graph max of s ----
  float lmax = -INFINITY;
  for (int i = tid; i < n; i += BLK) {
    const float4 c0 = *(const float4*)(close + (size_t)(i0 + i) * CLOSE_DIM);
    const float4 c1 = *(const float4*)(close + (size_t)(i0 + i) * CLOSE_DIM + 4);
    float s = bias + c0.x * wc[0] + c0.y * wc[1] + c0.z * wc[2] + c0.w * wc[3]
                   + c1.x * wc[4] + c1.y * wc[5] + c1.z * wc[6] + c1.w * wc[7];
    lmax = fmaxf(lmax, s);
  }
  red[tid] = lmax;
  if (tid < 512 - BLK) red[BLK + tid] = -INFINITY;
  __syncthreads();
  for (int st = 256; st >= 1; st >>= 1) {
    if (tid < st) red[tid] = fmaxf(red[tid], red[tid + st]);
    __syncthreads();
  }
  const float smax = red[0];
  __syncthreads();

  // ---- pass 2: den + per-column max of e*x (single streaming pass over x) ----
  const int col    = tid % NEURON;  // 0..47  (fixed column per thread)
  const int rowoff = tid / NEURON;  // 0..7
  float denp = 0.0f;
  float m    = -INFINITY;

  for (int base = 0; base < n; base += TILE) {
    if (tid < TILE) {
      const int i = base + tid;
      float e = 0.0f;
      if (i < n) {
        const float4 c0 = *(const float4*)(close + (size_t)(i0 + i) * CLOSE_DIM);
        const float4 c1 = *(const float4*)(close + (size_t)(i0 + i) * CLOSE_DIM + 4);
        float s = bias + c0.x * wc[0] + c0.y * wc[1] + c0.z * wc[2] + c0.w * wc[3]
                       + c1.x * wc[4] + c1.y * wc[5] + c1.z * wc[6] + c1.w * wc[7];
        e = __expf(s - smax);
        denp += e;
      }
      etile[tid] = e;
      // prefetch next tile of x (gfx1250 global_prefetch_b8)
      const int pn = base + TILE + tid;
      if (pn < n) __builtin_prefetch(x + (size_t)(i0 + pn) * NEURON, 0, 0);
    }
    __syncthreads();
#pragma unroll
    for (int j = 0; j < TILE / 8; ++j) {
      const int nit = rowoff + 8 * j;
      const int i   = base + nit;
      if (i < n) {
        const float xv = x[(size_t)(i0 + i) * NEURON + col];
        m = fmaxf(m, etile[nit] * xv);
      }
    }
    __syncthreads();
  }

  // reduce den (threads >= TILE contribute 0)
  red[tid] = denp;
  if (tid < 512 - BLK) red[BLK + tid] = 0.0f;
  __syncthreads();
  for (int st = 256; st >= 1; st >>= 1) {
    if (tid < st) red[tid] += red[tid + st];
    __syncthreads();
  }
  const float den = red[0];
  __syncthreads();

  // reduce per-column maxima across the 8 row-offset threads
  red[tid] = m;
  __syncthreads();
  if (tid < NEURON) {
    float mm = red[tid];
#pragma unroll
    for (int r = 1; r < 8; ++r) mm = fmaxf(mm, red[tid + NEURON * r]);
    pooled[g * NEURON + tid] = ((float)n / den) * mm;
  }
}

// ---------------------------------------------------------------------------
// Kernel 3: MLP head via V_WMMA_F32_16X16X4_F32.
// One wave per 16-row tile of pooled: h = relu(pooled@W1 + b1); out = h@W2 + b2.
// A 16x4 f32 frag: lane%16 = M; VGPR pair holds K = {0,1} (lanes 0-15) / {2,3}.
// B 4x16 f32 frag: lane%16 = N; VGPR v holds rows {v, v+2} split by lane half.
// C/D 16x16 f32: VGPR v -> row v + 8*(lane/16), col = lane%16.
// ---------------------------------------------------------------------------
__global__ __launch_bounds__(32)
void mlp_head_kernel(const float* __restrict__ pooled,
                     const float* __restrict__ W1,   // [48,16]
                     const float* __restrict__ b1,   // [16]
                     const float* __restrict__ W2,   // [16]
                     const float* __restrict__ b2,   // [1]
                     float* __restrict__ out) {      // [G]
  __shared__ float hsm[16 * 16];
  const int lane = threadIdx.x;
  const int half = lane >> 4;   // 0 or 1
  const int l16  = lane & 15;
  const int rowA = blockIdx.x * 16 + l16;   // M index of this lane's A rows

  v8f acc;
#pragma unroll
  for (int v = 0; v < 8; ++v) acc[v] = b1[l16];   // bias broadcast along N

#pragma unroll
  for (int kk = 0; kk < NEURON / 4; ++kk) {
    const int k0 = kk * 4;
    v2f a, b;
    a.x = pooled[(size_t)rowA * NEURON + k0 + 2 * half + 0];
    a.y = pooled[(size_t)rowA * NEURON + k0 + 2 * half + 1];
    b.x = W1[(k0 + 2 * half + 0) * HID + l16];
    b.y = W1[(k0 + 2 * half + 1) * HID + l16];
    acc = __builtin_amdgcn_wmma_f32_16x16x4_f32(
        /*neg_a=*/false, a, /*neg_b=*/false, b,
        /*c_mod=*/(short)0, acc, /*reuse_a=*/false, /*reuse_b=*/false);
  }

  // relu + spill h[16][16] to LDS using the documented C/D layout
#pragma unroll
  for (int v = 0; v < 8; ++v) {
    float h = acc[v];
    h = h > 0.0f ? h : 0.0f;
    hsm[(v + 8 * half) * 16 + l16] = h;
  }
  __syncthreads();

  if (lane < 16) {
    float a2 = b2[0];
#pragma unroll
    for (int c = 0; c < HID; ++c) a2 += hsm[lane * 16 + c] * W2[c];
    out[blockIdx.x * 16 + lane] = a2;
  }
}

// ---------------------------------------------------------------------------
// Host-side launcher
// inputs: x, closeness_feature, batch, Wc, bc, W1, b1, W2, b2, num_graphs
// ---------------------------------------------------------------------------
extern "C" void kernel_launch(void* const* d_in, const int* in_sizes, int n_in,
                              void* d_out, int out_size, void* d_ws, size_t ws_size,
                              hipStream_t stream) {
  const float* x     = (const float*)d_in[0];
  const float* close = (const float*)d_in[1];
  const int*   batch = (const int*)d_in[2];
  const float* Wc    = (const float*)d_in[3];
  const float* bc    = (const float*)d_in[4];
  const float* W1    = (const float*)d_in[5];
  const float* b1    = (const float*)d_in[6];
  const float* W2    = (const float*)d_in[7];
  const float* b2    = (const float*)d_in[8];
  float* out = (float*)d_out;

  const int n_nodes = in_sizes[2];

  // workspace layout: [0, 8192): int start[G+1]; [8192, ...): float pooled[G*48]
  int*   start  = (int*)d_ws;
  float* pooled = (float*)((char*)d_ws + 8192);

  seg_bounds_kernel<<<(N_GRAPHS + 1 + 255) / 256, 256, 0, stream>>>(batch, n_nodes, start);
  attn_pool_kernel<<<N_GRAPHS, BLK, 0, stream>>>(x, close, start, Wc, bc, pooled);
  mlp_head_kernel<<<N_GRAPHS / 16, 32, 0, stream>>>(pooled, W1, b1, W2, b2, out);
}